// Decoder_7816840479047
// MI455X (gfx1250) — compile-verified
//
#include <hip/hip_runtime.h>
#include <hip/hip_bf16.h>
#include <math.h>

// ---------------------------------------------------------------------------
// Fused ordinal-MLP decoder for MI455X (gfx1250, wave32, WMMA f32 16x16x4).
// x(64x512) strip -> h1 chunks -> h (register-resident) -> heads -> tail.
// Weight tiles stream L2 -> LDS via GLOBAL_LOAD_ASYNC_TO_LDS_B128 with a
// double-buffered ASYNCcnt pipeline (fallback: synchronous copy).
// ---------------------------------------------------------------------------

typedef __attribute__((ext_vector_type(2))) float v2f;
typedef __attribute__((ext_vector_type(8))) float v8f;
typedef int v4i_ __attribute__((vector_size(16)));   // matches builtin pointee

#define NB        65536      // batch rows
#define D_INP     512
#define H1DIM     1024
#define H2DIM     512
#define NITEMS    16
#define NCOLS     128        // 16 (f_x) + 16*7 (cut)
#define TPB       512        // 16 waves
#define BM        64         // batch rows per block

#define XS_STRIDE 516        // 512 + 4 pad  (bank advance 4)
#define HC_STRIDE 132        // 128 + 4 pad
#define WS_STRIDE 520        // 512 + 8 pad  (B-frag lane halves hit disjoint banks)

#if defined(__has_builtin)
#if __has_builtin(__builtin_amdgcn_global_load_async_to_lds_b128)
#define HAVE_ASYNC_COPY 1
#endif
#endif

// 16-byte global -> LDS copy (async on CDNA5, no VGPR round trip)
__device__ __forceinline__ void async_cp16(float* lds_dst, const float* gsrc) {
#ifdef HAVE_ASYNC_COPY
  __builtin_amdgcn_global_load_async_to_lds_b128(
      (__attribute__((address_space(1))) v4i_*)gsrc,
      (__attribute__((address_space(3))) v4i_*)lds_dst, 0, 0);
#else
  float4 v = *reinterpret_cast<const float4*>(gsrc);
  lds_dst[0] = v.x; lds_dst[1] = v.y; lds_dst[2] = v.z; lds_dst[3] = v.w;
#endif
}

__device__ __forceinline__ void async_join() {
#ifdef HAVE_ASYNC_COPY
#if __has_builtin(__builtin_amdgcn_s_wait_asynccnt)
  __builtin_amdgcn_s_wait_asynccnt(0);
#else
  asm volatile("s_wait_asynccnt 0x0" ::: "memory");
#endif
#endif
}

__device__ __forceinline__ v8f wmma_f32(v2f a, v2f b, v8f c) {
  // D = A(16x4) * B(4x16) + C(16x16), fp32, wave32
  return __builtin_amdgcn_wmma_f32_16x16x4_f32(false, a, false, b, (short)0, c,
                                               false, false);
}

// ---------------------------------------------------------------------------
// Prep: pack [Wf | Wk^T] into Wc (512 x 128), [bf | bk] into cb (128),
// zero the loss accumulator at out[0].
// ---------------------------------------------------------------------------
__global__ void prep_kernel(const float* __restrict__ Wf, const float* __restrict__ bf,
                            const float* __restrict__ Wk, const float* __restrict__ bk,
                            float* __restrict__ Wc, float* __restrict__ cb,
                            float* __restrict__ out0) {
  int gid = blockIdx.x * blockDim.x + threadIdx.x;   // 0 .. 65535
  if (gid == 0) out0[0] = 0.0f;
  if (gid < NCOLS) cb[gid] = (gid < NITEMS) ? bf[gid] : bk[gid - NITEMS];
  int k = gid >> 7;          // 0..511
  int n = gid & 127;         // 0..127
  float v;
  if (n < NITEMS) {
    v = Wf[k * NITEMS + n];                       // Wf is (512,16)
  } else {
    int m = n - NITEMS;                           // i*7 + c
    int i = m / 7, c = m - 7 * i;
    v = Wk[((size_t)i * H2DIM + k) * 7 + c];      // Wk is (16,512,7)
  }
  Wc[gid] = v;
}

// ---------------------------------------------------------------------------
// Fused megakernel. Grid = NB/BM = 1024 blocks, 512 threads (16 waves).
// ---------------------------------------------------------------------------
__global__ __launch_bounds__(TPB, 1)
void fused_decoder(const float* __restrict__ x, const int* __restrict__ data,
                   const float* __restrict__ W1, const float* __restrict__ b1,
                   const float* __restrict__ W2, const float* __restrict__ b2,
                   const float* __restrict__ Wc, const float* __restrict__ cb,
                   float* __restrict__ out) {
  __shared__ float xs[BM * XS_STRIDE];       // x strip; reused as h strip (132KB)
  __shared__ float h1c[BM * HC_STRIDE];      // h1 chunk; reused as logits (33KB)
  __shared__ float Ws[2][16 * WS_STRIDE];    // double-buffered weight stage (66KB)

  const int tid  = threadIdx.x;
  const int wid  = tid >> 5;          // wave 0..15
  const int lane = tid & 31;
  const int lr   = lane & 15;         // lane % 16
  const int lh   = lane >> 4;         // lane half (0/1)
  const int wr   = wid & 3;           // wave row-tile: rows wr*16 .. +15
  const int wc   = wid >> 2;          // wave col-group 0..3
  const int row0 = blockIdx.x * BM;

  // ---- async-load x strip (64 x 512) straight into LDS ----
#pragma unroll
  for (int i = 0; i < 16; ++i) {
    int idx = tid + i * TPB;              // 0..8191 float4 slots
    int r = idx >> 7, c4 = idx & 127;
    async_cp16(&xs[r * XS_STRIDE + c4 * 4],
               x + (size_t)(row0 + r) * D_INP + c4 * 4);
  }

  v8f hacc[8];
#pragma unroll
  for (int i = 0; i < 8; ++i) hacc[i] = (v8f){0,0,0,0,0,0,0,0};

  async_join();
  __syncthreads();

  const int sk = tid >> 5;      // staging row for 16x128 tiles (0..15)
  const int sn4 = tid & 31;     // staging float4 col

  // ================= fused GEMM1+GEMM2 over K-chunks of h1 =================
  for (int kc = 0; kc < 8; ++kc) {           // h1 columns kc*128 .. +127
    // ---- stage 1: h1c = relu(xs @ W1[:, chunk] + b1[chunk]) ----
    v8f acc1[2];
    acc1[0] = (v8f){0,0,0,0,0,0,0,0};
    acc1[1] = (v8f){0,0,0,0,0,0,0,0};
    // prologue: tile ks=0 -> Ws[0]
    async_cp16(&Ws[0][sk * WS_STRIDE + sn4 * 4],
               W1 + (size_t)sk * H1DIM + kc * NCOLS + sn4 * 4);
    async_join();
    __syncthreads();
    for (int ks = 0; ks < 32; ++ks) {        // K = 512 in stages of 16
      int cur = ks & 1;
      if (ks + 1 < 32)                        // issue next tile into other buffer
        async_cp16(&Ws[cur ^ 1][sk * WS_STRIDE + sn4 * 4],
                   W1 + (size_t)((ks + 1) * 16 + sk) * H1DIM + kc * NCOLS + sn4 * 4);
#pragma unroll
      for (int kk = 0; kk < 4; ++kk) {
        int kb = kk * 4 + lh * 2;
        v2f a = *reinterpret_cast<const v2f*>(
            &xs[(wr * 16 + lr) * XS_STRIDE + ks * 16 + kb]);
#pragma unroll
        for (int ct = 0; ct < 2; ++ct) {
          int n0 = wc * 32 + ct * 16;
          v2f b;
          b.x = Ws[cur][(kb + 0) * WS_STRIDE + n0 + lr];
          b.y = Ws[cur][(kb + 1) * WS_STRIDE + n0 + lr];
          acc1[ct] = wmma_f32(a, b, acc1[ct]);
        }
      }
      if (ks + 1 < 32) async_join();
      __syncthreads();
    }
    // epilogue: bias + relu -> h1c
#pragma unroll
    for (int ct = 0; ct < 2; ++ct) {
      int n0 = wc * 32 + ct * 16;
      float bias = b1[kc * NCOLS + n0 + lr];
#pragma unroll
      for (int j = 0; j < 8; ++j) {
        int r = wr * 16 + lh * 8 + j;        // C/D layout: vgpr j -> M=j / j+8
        h1c[r * HC_STRIDE + n0 + lr] = fmaxf(acc1[ct][j] + bias, 0.0f);
      }
    }
    __syncthreads();

    // ---- stage 2: hacc += h1c @ W2[chunk, :] ----
    // prologue: tile ks=0 (16x512) -> Ws[0], 4 float4 per thread
#pragma unroll
    for (int i = 0; i < 4; ++i) {
      int idx = tid + i * TPB;
      int k = idx >> 7, n4 = idx & 127;
      async_cp16(&Ws[0][k * WS_STRIDE + n4 * 4],
                 W2 + (size_t)(kc * 128 + k) * H2DIM + n4 * 4);
    }
    async_join();
    __syncthreads();
    for (int ks = 0; ks < 8; ++ks) {         // chunk K = 128 in stages of 16
      int cur = ks & 1;
      if (ks + 1 < 8) {
#pragma unroll
        for (int i = 0; i < 4; ++i) {
          int idx = tid + i * TPB;
          int k = idx >> 7, n4 = idx & 127;
          async_cp16(&Ws[cur ^ 1][k * WS_STRIDE + n4 * 4],
                     W2 + (size_t)(kc * 128 + (ks + 1) * 16 + k) * H2DIM + n4 * 4);
        }
      }
#pragma unroll
      for (int kk = 0; kk < 4; ++kk) {
        int kb = kk * 4 + lh * 2;
        v2f a = *reinterpret_cast<const v2f*>(
            &h1c[(wr * 16 + lr) * HC_STRIDE + ks * 16 + kb]);
#pragma unroll
        for (int ct = 0; ct < 8; ++ct) {
          int n0 = wc * 128 + ct * 16;
          v2f b;
          b.x = Ws[cur][(kb + 0) * WS_STRIDE + n0 + lr];
          b.y = Ws[cur][(kb + 1) * WS_STRIDE + n0 + lr];
          hacc[ct] = wmma_f32(a, b, hacc[ct]);
        }
      }
      if (ks + 1 < 8) async_join();
      __syncthreads();
    }
  }

  // ---- h = relu(hacc + b2) -> xs (x strip is dead now) ----
#pragma unroll
  for (int ct = 0; ct < 8; ++ct) {
    int col = wc * 128 + ct * 16 + lr;
    float bias = b2[col];
#pragma unroll
    for (int j = 0; j < 8; ++j) {
      int r = wr * 16 + lh * 8 + j;
      xs[r * XS_STRIDE + col] = fmaxf(hacc[ct][j] + bias, 0.0f);
    }
  }
  __syncthreads();

  // ---- stage 3: logits(64x128) = h @ Wc + cb -> h1c ----
  v8f acc3[2];
  acc3[0] = (v8f){0,0,0,0,0,0,0,0};
  acc3[1] = (v8f){0,0,0,0,0,0,0,0};
  async_cp16(&Ws[0][sk * WS_STRIDE + sn4 * 4],
             Wc + (size_t)sk * NCOLS + sn4 * 4);
  async_join();
  __syncthreads();
  for (int ks = 0; ks < 32; ++ks) {
    int cur = ks & 1;
    if (ks + 1 < 32)
      async_cp16(&Ws[cur ^ 1][sk * WS_STRIDE + sn4 * 4],
                 Wc + (size_t)((ks + 1) * 16 + sk) * NCOLS + sn4 * 4);
#pragma unroll
    for (int kk = 0; kk < 4; ++kk) {
      int kb = kk * 4 + lh * 2;
      v2f a = *reinterpret_cast<const v2f*>(
          &xs[(wr * 16 + lr) * XS_STRIDE + ks * 16 + kb]);
#pragma unroll
      for (int ct = 0; ct < 2; ++ct) {
        int n0 = wc * 32 + ct * 16;
        v2f b;
        b.x = Ws[cur][(kb + 0) * WS_STRIDE + n0 + lr];
        b.y = Ws[cur][(kb + 1) * WS_STRIDE + n0 + lr];
        acc3[ct] = wmma_f32(a, b, acc3[ct]);
      }
    }
    if (ks + 1 < 32) async_join();
    __syncthreads();
  }
#pragma unroll
  for (int ct = 0; ct < 2; ++ct) {
    int n0 = wc * 32 + ct * 16;
    float bias = cb[n0 + lr];
#pragma unroll
    for (int j = 0; j < 8; ++j) {
      int r = wr * 16 + lh * 8 + j;
      h1c[r * HC_STRIDE + n0 + lr] = acc3[ct][j] + bias;
    }
  }
  __syncthreads();

  // ---- ordinal tail: 64 rows * 16 items / 512 threads = 2 each ----
  const float EPS = 1e-8f;
  float lsum = 0.0f;
#pragma unroll
  for (int p = 0; p < 2; ++p) {
    int pi = tid + p * TPB;                  // 0..1023
    int r = pi >> 4, item = pi & 15;
    const float* L = &h1c[r * HC_STRIDE];
    float fx = L[item];
    int cp_n = 4 + (item & 3);               // NUM_CUTS = [4,5,6,7]*4
    float probs[8];
    float cum = 0.0f, prev = 0.0f, last = 0.0f;
#pragma unroll
    for (int c = 0; c < 7; ++c) {
      float cutv = L[NITEMS + item * 7 + c];
      float e = (c < cp_n) ? fmaxf(expf(cutv), EPS) : 0.0f;
      cum += e;
      float cpv = 1.0f / (1.0f + expf(-(cum - fx)));
      cpv = fminf(fmaxf(cpv, EPS), 1.0f - EPS);
      float d = (c == 0) ? cpv : (cpv - prev);
      prev = cpv;
      if (c < cp_n) probs[c] = d;
      if (c == cp_n - 1) last = cpv;
    }
    probs[cp_n] = 1.0f - last;
    for (int c = cp_n + 1; c < 8; ++c) probs[c] = 1.0f;
#pragma unroll
    for (int c = 0; c < 8; ++c)
      probs[c] = fminf(fmaxf(probs[c], EPS), 1.0f - EPS);

    size_t gb = (size_t)(row0 + r) * NITEMS + item;
    float* po = out + 1 + gb * 8;
#pragma unroll
    for (int c = 0; c < 8; ++c) po[c] = probs[c];
    int dv = data[gb];                        // 0..3
    lsum -= logf(probs[dv]);
  }
  // wave reduce then one atomic per wave
#pragma unroll
  for (int off = 16; off > 0; off >>= 1) lsum += __shfl_down(lsum, off, 32);
  if (lane == 0) atomicAdd(out, lsum * (1.0f / (float)NB));
}

// ---------------------------------------------------------------------------
extern "C" void kernel_launch(void* const* d_in, const int* in_sizes, int n_in,
                              void* d_out, int out_size, void* d_ws, size_t ws_size,
                              hipStream_t stream) {
  (void)in_sizes; (void)n_in; (void)out_size; (void)ws_size;
  const float* x    = (const float*)d_in[0];
  const int*   data = (const int*)d_in[1];
  const float* W1   = (const float*)d_in[2];
  const float* b1   = (const float*)d_in[3];
  const float* W2   = (const float*)d_in[4];
  const float* b2   = (const float*)d_in[5];
  const float* Wf   = (const float*)d_in[6];
  const float* bf   = (const float*)d_in[7];
  const float* Wk   = (const float*)d_in[8];
  const float* bk   = (const float*)d_in[9];
  float* out = (float*)d_out;

  float* Wc = (float*)d_ws;                 // 512*128 floats = 256KB
  float* cb = Wc + D_INP * NCOLS;           // 128 floats

  prep_kernel<<<(D_INP * NCOLS) / 256, 256, 0, stream>>>(Wf, bf, Wk, bk, Wc, cb, out);
  fused_decoder<<<NB / BM, TPB, 0, stream>>>(x, data, W1, b1, W2, b2, Wc, cb, out);
}